// DenseBlockDiagAPDAGDLayer_18047452578724
// MI455X (gfx1250) — compile-verified
//
#include <hip/hip_runtime.h>
#include <math.h>

// APDAGD block-diagonal LP solver for MI455X (gfx1250).
// One persistent workgroup per diagonal block; A_k staged in LDS (256KB fits in
// the 320KB WGP LDS) via async global->LDS loads; A^T matvecs via
// V_WMMA_F32_16X16X4_F32 in full f32.

typedef __attribute__((ext_vector_type(2))) float v2f;
typedef __attribute__((ext_vector_type(8))) float v8f;
typedef int v4i __attribute__((vector_size(16)));           // matches builtin param
typedef __attribute__((address_space(1))) v4i gv4i;          // global int4
typedef __attribute__((address_space(3))) v4i lv4i;          // LDS int4

#define NKB   32          // diagonal blocks
#define NC    64          // constraints per block
#define NV    1024        // variables per block
#define ROWW  (NKB * NV)  // global A row width = 32768
#define AST   1032        // padded LDS row stride (words): 2*1032 % 64 = 16 -> halves hit disjoint banks
#define NTHR  1024        // threads per workgroup (== NV), 32 wave32s

// ---- LDS layout (floats) ----
#define OFF_A     0
#define SZ_A      (NC * AST)          // 66048
#define OFF_U     (OFF_A + SZ_A)
#define OFF_TU    (OFF_U + NV)
#define OFF_C     (OFF_TU + NV)
#define OFF_XPU   (OFF_C + NV)
#define OFF_SL    (OFF_XPU + NV)
#define OFF_W     (OFF_SL + NV)
#define OFF_B     (OFF_W + NV)
#define OFF_ETA   (OFF_B + NC)
#define OFF_ZETA  (OFF_ETA + NC)
#define OFF_LAM   (OFF_ZETA + NC)
#define OFF_ETAN  (OFF_LAM + NC)
#define OFF_ZNEW  (OFF_ETAN + NC)
#define OFF_AX    (OFF_ZNEW + NC)
#define OFF_RED   (OFF_AX + NC)
#define OFF_SCAL  (OFF_RED + 32)
#define SMEM_FLOATS (OFF_SCAL + 16)   // ~290.8 KB total < 320 KB

// scal slots
#define S_M     0
#define S_BETA  1
#define S_LASTC 2
#define S_BTB   3
#define S_CONT  5

__device__ __forceinline__ float softplus_f(float x) {
  // logaddexp(0, x), numerically stable
  return fmaxf(x, 0.f) + log1pf(expf(-fabsf(x)));
}

__device__ __forceinline__ float wave_sum(float v) {
#pragma unroll
  for (int off = 16; off > 0; off >>= 1) v += __shfl_xor(v, off, 32);
  return v;
}

__device__ __forceinline__ float block_sum(float v, float* s_red, int tid) {
  v = wave_sum(v);
  __syncthreads();                       // protect s_red from previous use
  if ((tid & 31) == 0) s_red[tid >> 5] = v;
  __syncthreads();
  if (tid < 32) {
    float r = s_red[tid];
    r = wave_sum(r);
    if (tid == 0) s_red[0] = r;
  }
  __syncthreads();
  return s_red[0];
}

// out[1024] = A^T(64x1024 in LDS, row stride AST) * vec[64], via WMMA f32 16x16x4.
// A-operand layout (ISA 7.12.2, 32-bit A 16x4): lanes 0-15 hold M=0..15 with
// K=0 (v0), K=1 (v1); lanes 16-31 hold K=2 (v0), K=3 (v1).
// B-operand (4x16, rows striped across lane halves per VGPR): every column is
// the same lam chunk, so every column of D equals the matvec chunk.
__device__ __forceinline__ void atmv_wmma(const float* sA, const float* vec,
                                          float* out, int tid) {
  const int lane = tid & 31;
  const int wave = tid >> 5;
  const int half = lane >> 4;   // 0: K=0,1  1: K=2,3
  const int m    = lane & 15;   // matrix row within tile
#pragma unroll
  for (int tt = 0; tt < 2; ++tt) {
    const int t  = wave + tt * 32;      // variable tile 0..63
    const int vb = t * 16 + m;
    v8f acc = {0.f, 0.f, 0.f, 0.f, 0.f, 0.f, 0.f, 0.f};
#pragma unroll
    for (int kb = 0; kb < NC; kb += 4) {
      const int k0 = kb + (half << 1);
      v2f a, b;
      a.x = sA[k0 * AST + vb];
      a.y = sA[(k0 + 1) * AST + vb];
      b.x = vec[kb + half];
      b.y = vec[kb + 2 + half];
      acc = __builtin_amdgcn_wmma_f32_16x16x4_f32(
          /*neg_a=*/false, a, /*neg_b=*/false, b,
          /*c_mod=*/(short)0, acc, /*reuse_a=*/false, /*reuse_b=*/false);
    }
    // D: lane (n + 16*(M>=8)), vgpr M%8. Column n=0 lives in lanes 0 (M=0..7)
    // and 16 (M=8..15); all columns identical, so those two lanes write out.
    if (m == 0) {
      const int rb = t * 16 + half * 8;
#pragma unroll
      for (int r = 0; r < 8; ++r) out[rb + r] = acc[r];
    }
  }
}

// outAx[64] = A(64x1024 in LDS) * w[1024], lane-parallel FMA + LDS atomics.
__device__ __forceinline__ void amv(const float* sA, const float* w,
                                    float* outAx, int tid) {
  if (tid < NC) outAx[tid] = 0.f;
  __syncthreads();
  const int c  = tid >> 4;
  const int s0 = tid & 15;
  const float* row = sA + c * AST;
  float p = 0.f;
#pragma unroll 8
  for (int j = 0; j < 64; ++j) p += row[s0 + 16 * j] * w[s0 + 16 * j];
  atomicAdd(&outAx[c], p);
  __syncthreads();
}

__global__ void init_sync_kernel(int* g) {
  const int i = threadIdx.x + blockIdx.x * blockDim.x;
  if (i < 512) g[i] = 0;
}

__global__ void __launch_bounds__(NTHR, 1)
apdagd_kernel(const float* __restrict__ gA, const float* __restrict__ gb,
              const float* __restrict__ gc, const float* __restrict__ gu,
              const float* __restrict__ theta_p, const float* __restrict__ eps_p,
              const int* __restrict__ maxit_p, float* __restrict__ d_out,
              int* __restrict__ g_sync) {
  extern __shared__ float sm[];
  float* sA     = sm + OFF_A;
  float* s_u    = sm + OFF_U;
  float* s_tu   = sm + OFF_TU;
  float* s_c    = sm + OFF_C;
  float* s_xpu  = sm + OFF_XPU;
  float* s_sl   = sm + OFF_SL;
  float* s_w    = sm + OFF_W;
  float* s_b    = sm + OFF_B;
  float* s_eta  = sm + OFF_ETA;
  float* s_zeta = sm + OFF_ZETA;
  float* s_lam  = sm + OFF_LAM;
  float* s_etan = sm + OFF_ETAN;
  float* s_znew = sm + OFF_ZNEW;
  float* s_Ax   = sm + OFF_AX;
  float* s_red  = sm + OFF_RED;
  float* s_scal = sm + OFF_SCAL;

  const int k   = blockIdx.x;
  const int tid = threadIdx.x;
  const float theta = theta_p[0];
  const float epsv  = eps_p[0];
  int max_iter = maxit_p[0];
  if (max_iter > 255) max_iter = 255;   // ws sync slots capacity
  const float deps = 1.1920929e-07f;    // float32 machine eps

  // ---- Stage A diagonal block (rows 64k.., cols 1024k..) into LDS ----
  const float* Ablk = gA + (size_t)(k * NC) * ROWW + (size_t)k * NV;
#if __has_builtin(__builtin_amdgcn_global_load_async_to_lds_b128)
  for (int idx = tid; idx < NC * 256; idx += NTHR) {   // 16384 x 16B
    const int row = idx >> 8;
    const int q   = (idx & 255) << 2;                  // float offset in row
    __builtin_amdgcn_global_load_async_to_lds_b128(
        (gv4i*)(Ablk + (size_t)row * ROWW + q),
        (lv4i*)(sA + row * AST + q), 0, 0);
  }
#if __has_builtin(__builtin_amdgcn_s_wait_asynccnt)
  __builtin_amdgcn_s_wait_asynccnt(0);
#else
  asm volatile("s_wait_asynccnt 0" ::: "memory");
#endif
#else
  for (int idx = tid; idx < NC * 256; idx += NTHR) {
    const int row = idx >> 8;
    const int q   = (idx & 255) << 2;
    const float4 v = *(const float4*)(Ablk + (size_t)row * ROWW + q);
    float* dst = sA + row * AST + q;
    dst[0] = v.x; dst[1] = v.y; dst[2] = v.z; dst[3] = v.w;
  }
#endif

  // ---- Stage vectors / init state ----
  {
    const float uu = gu[(size_t)k * NV + tid];
    s_u[tid]  = uu;
    s_tu[tid] = theta * uu;
    s_c[tid]  = gc[(size_t)k * NV + tid];
  }
  if (tid < NC) {
    s_b[tid]    = gb[k * NC + tid];
    s_eta[tid]  = 0.f;
    s_zeta[tid] = 0.f;
  }
  __syncthreads();
  if (tid == 0) {
    float bt = 0.f;
    for (int i = 0; i < NC; ++i) bt += s_b[i] * s_b[i];
    s_scal[S_BTB]   = bt;
    s_scal[S_M]     = theta;
    s_scal[S_BETA]  = 0.f;
    s_scal[S_LASTC] = 0.f;
  }
  // x_pu0 = sigmoid(-c * theta_u)  (eta0 == 0)
  s_xpu[tid] = 1.f / (1.f + expf(s_c[tid] * s_tu[tid]));
  __syncthreads();

  // ---- do-while iteration loop (globally synchronized stop test) ----
  int it = 0;
  while (true) {
    const float M     = s_scal[S_M];
    const float beta  = s_scal[S_BETA];
    const bool  lastc = (s_scal[S_LASTC] != 0.f);
    const float alpha  = 0.5f / M + sqrtf((0.25f / M + beta) / M);
    const float beta_n = beta + alpha;
    const float tau    = alpha / beta_n;

    if (tid < NC) s_lam[tid] = s_eta[tid] + tau * (s_zeta[tid] - s_eta[tid]);
    __syncthreads();

    // s_l = -(c - A^T lam) * theta_u
    atmv_wmma(sA, s_lam, s_sl, tid);
    __syncthreads();
    const float sl = -(s_c[tid] - s_sl[tid]) * s_tu[tid];
    s_sl[tid] = sl;
    const float xl = 1.f / (1.f + expf(-sl));
    s_w[tid] = s_u[tid] * xl;              // u * x_l
    __syncthreads();

    // Ax = A (u * x_l)
    amv(sA, s_w, s_Ax, tid);

    if (tid < NC) {
      const float g  = s_Ax[tid] - s_b[tid];
      const float zn = s_zeta[tid] - alpha * g;
      s_znew[tid] = zn;
      s_etan[tid] = s_eta[tid] + tau * (zn - s_eta[tid]);
    }
    __syncthreads();

    // raw A^T eta_n -> s_w (u*x_l no longer needed)
    atmv_wmma(sA, s_etan, s_w, tid);
    __syncthreads();

    // cond = (sum(Ax^2)-btb)*0.5/M + sum((sp(s_e)-sp(s_l))/theta) <= eps_f32
    const float se = -(s_c[tid] - s_w[tid]) * s_tu[tid];
    float val = (softplus_f(se) - softplus_f(sl)) / theta;
    if (tid < NC) { const float ax = s_Ax[tid]; val += ax * ax * (0.5f / M); }
    if (tid == 0) val -= s_scal[S_BTB] * (0.5f / M);
    const float tot  = block_sum(val, s_red, tid);
    const bool  cond = (tot <= deps);

    if (tid == 0) {
      s_scal[S_M]     = fmaxf(cond ? (lastc ? M * 0.5f : M) : M * 2.0f, deps);
      s_scal[S_BETA]  = cond ? beta_n : beta;
      s_scal[S_LASTC] = cond ? 1.f : 0.f;
    }
    if (cond) {
      if (tid < NC) { s_eta[tid] = s_etan[tid]; s_zeta[tid] = s_znew[tid]; }
      s_xpu[tid] += tau * (xl - s_xpu[tid]);
    }
    __syncthreads();

    // pinf = ||A (u*x_pu) - b||
    s_w[tid] = s_u[tid] * s_xpu[tid];
    __syncthreads();
    amv(sA, s_w, s_Ax, tid);
    float pv = 0.f;
    if (tid < NC) { const float d = s_Ax[tid] - s_b[tid]; pv = d * d; }
    const float p2   = block_sum(pv, s_red, tid);
    const bool  conv = (sqrtf(p2) <= epsv);

    ++it;
    // ---- grid-wide convergence vote (nc flags [0..255], arrivals [256..511]) ----
    if (tid == 0) {
      if (!conv) atomicOr(&g_sync[it], 1);
      __hip_atomic_fetch_add(&g_sync[256 + it], 1, __ATOMIC_RELEASE,
                             __HIP_MEMORY_SCOPE_AGENT);
      int a;
      do {
        a = __hip_atomic_load(&g_sync[256 + it], __ATOMIC_ACQUIRE,
                              __HIP_MEMORY_SCOPE_AGENT);
      } while (a < NKB);
      const int nc = __hip_atomic_load(&g_sync[it], __ATOMIC_RELAXED,
                                       __HIP_MEMORY_SCOPE_AGENT);
      s_scal[S_CONT] = ((it < max_iter) && (nc != 0)) ? 1.f : 0.f;
    }
    __syncthreads();
    if (s_scal[S_CONT] == 0.f) break;
  }

  // ---- outputs: [u*x_pu (32768)] ++ [eta (2048)] ----
  d_out[(size_t)k * NV + tid] = s_u[tid] * s_xpu[tid];
  if (tid < NC) d_out[(size_t)NKB * NV + k * NC + tid] = s_eta[tid];
}

extern "C" void kernel_launch(void* const* d_in, const int* in_sizes, int n_in,
                              void* d_out, int out_size, void* d_ws, size_t ws_size,
                              hipStream_t stream) {
  (void)in_sizes; (void)n_in; (void)out_size; (void)ws_size;
  const float* A     = (const float*)d_in[0];
  const float* b     = (const float*)d_in[1];
  const float* c     = (const float*)d_in[2];
  const float* u     = (const float*)d_in[3];
  // d_in[4], d_in[5]: n_c / n_v (fixed sizes, unused)
  const float* theta = (const float*)d_in[6];
  const float* eps   = (const float*)d_in[7];
  const int*   maxit = (const int*)d_in[8];
  int* gs = (int*)d_ws;

  const size_t shmem = (size_t)SMEM_FLOATS * sizeof(float);
  (void)hipFuncSetAttribute(reinterpret_cast<const void*>(apdagd_kernel),
                            hipFuncAttributeMaxDynamicSharedMemorySize,
                            (int)shmem);

  init_sync_kernel<<<dim3(1), dim3(512), 0, stream>>>(gs);
  apdagd_kernel<<<dim3(NKB), dim3(NTHR), shmem, stream>>>(
      A, b, c, u, theta, eps, maxit, (float*)d_out, gs);
}